// HeadwiseLowRankModule_12378095747715
// MI455X (gfx1250) — compile-verified
//
#include <hip/hip_runtime.h>

// ---------------------------------------------------------------------------
// Fused HeadwiseLowRank:  out = concat_g( (X @ VT_g^T) @ U_g^T + b_g )
//   X:  [16384, 4096] fp32      VT: [2048, 4096] fp32
//   U:  [32, 128, 64] fp32      b:  [32, 128]    fp32
//   out:[16384, 4096] fp32
// Memory-bound (~570 MB min traffic vs ~275 GFLOP at 23.3 TB/s -> ~25 us):
//  - fuse both stages (latents never touch HBM)
//  - fp32 WMMA (V_WMMA_F32_16X16X4_F32): matrix pipes are nowhere near the
//    bottleneck, so keep full precision
//  - GLOBAL_LOAD_ASYNC_TO_LDS_B128 + ASYNCcnt double buffering for the
//    HBM->LDS stream (no VGPR round trip, loads overlap WMMA compute)
// ---------------------------------------------------------------------------

typedef __attribute__((ext_vector_type(2))) float v2f;
typedef __attribute__((ext_vector_type(8))) float v8f;

#define WMMA_F32_4(a, b, c) \
  __builtin_amdgcn_wmma_f32_16x16x4_f32(false, (a), false, (b), (short)0, (c), false, false)

namespace {
constexpr int IN   = 4096;   // stage-1 reduction dim
constexpr int ROWS = 16384;  // B*L
constexpr int G    = 32;     // groups
constexpr int R    = 64;     // rank per group (stage-2 K)
constexpr int GD   = 128;    // group output dim
constexpr int OUT  = G * GD; // 4096

constexpr int BM = 128;      // rows per block
constexpr int BN = 64;       // latent cols per block (== R)
constexpr int BK = 32;       // K chunk staged in LDS
constexpr int AS = BK + 4;   // padded LDS stride (dwords) for A/B tiles
constexpr int US = R + 4;    // padded LDS stride (dwords) for U / latent tiles
constexpr int NT = IN / BK;  // 128 K-chunks
} // namespace

__global__ __launch_bounds__(256)
void hlr_fused_wmma(const float* __restrict__ hidden,
                    const float* __restrict__ VTw,
                    const float* __restrict__ Uw,
                    const float* __restrict__ Ub,
                    float* __restrict__ out)
{
    __shared__ __align__(16) float Asm[2][BM * AS];   // 36 KB  X tile   [m][k]
    __shared__ __align__(16) float Bsm[2][BN * AS];   // 18 KB  VT tile  [n][k]
    __shared__ __align__(16) float Usm[GD * US];      // 34 KB  U_g      [d][r]
    __shared__ __align__(16) float Lsm[8 * 16 * US];  // 34 KB  latents  [wave*16+m][r]

    const int g    = blockIdx.x;           // group (fastest -> L2 reuse of X)
    const int rb   = blockIdx.y;           // row block
    const int t    = threadIdx.x;
    const int lane = t & 31;
    const int wave = t >> 5;
    const int l15  = lane & 15;
    const int kh   = lane >> 4;            // 0: K+0/1 half, 1: K+2/3 half
    const size_t rowBase = (size_t)rb * BM;

    // ---- preload U_w[g] (128 x 64) into LDS, B-fragment friendly [d][r] ----
    {
        const float4* Ug = (const float4*)(Uw + (size_t)g * GD * R);
        #pragma unroll
        for (int i = 0; i < 8; ++i) {
            int c  = t + i * 256;          // float4 chunk id, 2048 total
            int d  = c >> 4;               // 16 float4 per row of 64 floats
            int r4 = (c & 15) << 2;
            float4 v = Ug[c];
            float* dst = &Usm[d * US + r4];
            dst[0] = v.x; dst[1] = v.y; dst[2] = v.z; dst[3] = v.w;
        }
    }

    // Issue one K-chunk's HBM->LDS stream with async-to-LDS (16B per lane per
    // op, 6 ops per thread: 4 for the X tile, 2 for the VT tile).
    auto issue_tile = [&](int k0, int buf) {
        #pragma unroll
        for (int i = 0; i < 4; ++i) {                  // X: 128x32 fl = 1024 f4
            int c  = t + i * 256;
            int r  = c >> 3;                           // 8 float4 per row
            int kc = (c & 7) << 2;
            const float* src = hidden + (rowBase + r) * (size_t)IN + k0 + kc;
            unsigned lds = (unsigned)(uintptr_t)&Asm[buf][r * AS + kc];
            asm volatile("global_load_async_to_lds_b128 %0, %1, off"
                         :: "v"(lds), "v"(src) : "memory");
        }
        #pragma unroll
        for (int i = 0; i < 2; ++i) {                  // VT: 64x32 fl = 512 f4
            int c  = t + i * 256;
            int n  = c >> 3;
            int kc = (c & 7) << 2;
            const float* src = VTw + ((size_t)g * BN + n) * IN + k0 + kc;
            unsigned lds = (unsigned)(uintptr_t)&Bsm[buf][n * AS + kc];
            asm volatile("global_load_async_to_lds_b128 %0, %1, off"
                         :: "v"(lds), "v"(src) : "memory");
        }
    };

    // ---- stage 1: latent tile [wave*16 .. +16) x [0..64) over K = 4096 ----
    v8f acc[4] = {};                        // 4 N-tiles of 16x16 f32

    issue_tile(0, 0);
    for (int it = 0; it < NT; ++it) {
        // my async writes for tile `it` have landed ...
        asm volatile("s_wait_asynccnt 0x0" ::: "memory");
        // ... and everyone else's too; also: all waves finished reading the
        // buffer that tile `it+1` will overwrite (they read it before the
        // *current* barrier), so issuing after this barrier is WAR-safe.
        __syncthreads();
        if (it + 1 < NT) issue_tile((it + 1) * BK, (it + 1) & 1);

        const float* As = Asm[it & 1];
        const float* Bs = Bsm[it & 1];
        #pragma unroll
        for (int kk = 0; kk < BK; kk += 4) {
            // A frag: lanes 0-15 -> (M=l15, K=kk,kk+1); lanes 16-31 -> kk+2,kk+3
            v2f a = *(const v2f*)&As[(wave * 16 + l15) * AS + kk + 2 * kh];
            #pragma unroll
            for (int n = 0; n < 4; ++n) {
                v2f b = *(const v2f*)&Bs[(n * 16 + l15) * AS + kk + 2 * kh];
                acc[n] = WMMA_F32_4(a, b, acc[n]);
            }
        }
    }

    // ---- spill latent tile to LDS in A-fragment layout (wave-private) ----
    // C layout: VGPR v, lanes 0-15 -> M=v ; lanes 16-31 -> M=v+8 ; N = lane%16
    #pragma unroll
    for (int n = 0; n < 4; ++n)
        #pragma unroll
        for (int v = 0; v < 8; ++v)
            Lsm[(wave * 16 + v + 8 * kh) * US + n * 16 + l15] = acc[n][v];

    // ---- stage 2: out tile [16 x 128] = Lat(16x64) @ U_g^T + b_g ----
    const size_t outRow0 = rowBase + (size_t)wave * 16;
    #pragma unroll 1
    for (int dt = 0; dt < 8; ++dt) {       // 8 N-tiles of 16 output dims
        float bias = Ub[g * GD + dt * 16 + l15];
        v8f acc2;
        #pragma unroll
        for (int v = 0; v < 8; ++v) acc2[v] = bias;

        #pragma unroll
        for (int kk = 0; kk < R; kk += 4) {
            v2f a = *(const v2f*)&Lsm[(wave * 16 + l15) * US + kk + 2 * kh];
            v2f b = *(const v2f*)&Usm[(dt * 16 + l15) * US + kk + 2 * kh];
            acc2 = WMMA_F32_4(a, b, acc2);
        }

        #pragma unroll
        for (int v = 0; v < 8; ++v) {
            size_t row = outRow0 + v + 8 * kh;
            out[row * OUT + (size_t)g * GD + dt * 16 + l15] = acc2[v];
        }
    }
}

extern "C" void kernel_launch(void* const* d_in, const int* in_sizes, int n_in,
                              void* d_out, int out_size, void* d_ws, size_t ws_size,
                              hipStream_t stream) {
    (void)in_sizes; (void)n_in; (void)out_size; (void)d_ws; (void)ws_size;
    const float* hidden = (const float*)d_in[0];  // [4,4096,4096]
    const float* VTw    = (const float*)d_in[1];  // [2048,4096]
    const float* Uw     = (const float*)d_in[2];  // [32,128,64]
    const float* Ub     = (const float*)d_in[3];  // [32,128]
    float* out          = (float*)d_out;          // [4,4096,4096]

    dim3 grid(G, ROWS / BM);                      // (32, 128)
    hlr_fused_wmma<<<grid, 256, 0, stream>>>(hidden, VTw, Uw, Ub, out);
}